// GCNModel_32143535243970
// MI455X (gfx1250) — compile-verified
//
#include <hip/hip_runtime.h>
#include <math.h>

#define N_NODES  100000
#define N_EDGES  1600000
#define N_GRAPHS 256
#define IN_CH    128
#define HID      64
#define OUT_CH   10

typedef float v2f __attribute__((ext_vector_type(2)));
typedef float v8f __attribute__((ext_vector_type(8)));

// ---------------------------------------------------------------------------
// Degree count: deg[dst] += 1 per edge (deg buffer pre-zeroed)
// ---------------------------------------------------------------------------
__global__ __launch_bounds__(256) void deg_count(const int* __restrict__ dst,
                                                 float* __restrict__ deg) {
    int e = blockIdx.x * 256 + threadIdx.x;
    if (e < N_EDGES) atomicAdd(&deg[dst[e]], 1.0f);
}

// dinv[n] = rsqrt(deg[n] + 1)   (self loop); done in place
__global__ __launch_bounds__(256) void make_dinv(float* __restrict__ deg) {
    int n = blockIdx.x * 256 + threadIdx.x;
    if (n < N_NODES) deg[n] = rsqrtf(deg[n] + 1.0f);
}

// ---------------------------------------------------------------------------
// WMMA fp32 GEMM:  out[nRows x 64] = [A1 | A2] @ W (+ bias)
//   A1: nRows x K1, A2: nRows x K2 (K2 may be 0), W: (K1+K2) x 64 row major.
// K1/K2 are template params -> fully unrolled K-loop, immediate-offset loads.
// W is staged in LDS pair-interleaved: for row-pair p (rows 2p, 2p+1), col c:
//   sW[p*128 + 2c + 0] = W[2p][c], sW[p*128 + 2c + 1] = W[2p+1][c]
// so a lane's B fragment {W[k0+2h][c], W[k0+2h+1][c]} is ONE ds_load_b64.
// One wave -> 16x64 tile via 4x V_WMMA_F32_16X16X4_F32 per k-step;
// 8 waves / block -> 128 rows per block.
// ---------------------------------------------------------------------------
template <int K1, int K2>
__global__ __launch_bounds__(256) void gemm_wmma_f32(
    const float* __restrict__ A1, const float* __restrict__ A2,
    const float* __restrict__ Wg, const float* __restrict__ bias,
    float* __restrict__ out, int nRows)
{
    constexpr int K = K1 + K2;
    __shared__ float sW[K * HID];                   // max 128*64 = 32 KB

    const int tid = threadIdx.x;
    for (int i = tid; i < K * HID; i += 256) {      // pair-interleaving swizzle
        int k = i / HID, c = i - k * HID;
        sW[(k >> 1) * (2 * HID) + 2 * c + (k & 1)] = Wg[i];
    }
    __syncthreads();

    const int wave = tid >> 5;
    const int lane = tid & 31;
    const int m0   = blockIdx.x * 128 + wave * 16;
    if (m0 >= nRows) return;                        // wave-uniform: EXEC stays full

    const int mrow = m0 + (lane & 15);              // A-frag row (M = lane%16)
    const int half = lane >> 4;                     // 0: lanes 0-15, 1: lanes 16-31
    const int col  = lane & 15;                     // N within a 16-wide tile

    const float* pA1 = A1 + (size_t)mrow * K1 + 2 * half;
    const float* pA2 = (K2 > 0) ? (A2 + (size_t)mrow * K2 + 2 * half) : A1;
    const v2f*   sB  = (const v2f*)sW;              // index = p*64 + (n*16+col)
    const int    bix = half * HID + col;            // p-offset folded per lane

    v8f acc0 = {}, acc1 = {}, acc2 = {}, acc3 = {};

    #pragma unroll
    for (int k0 = 0; k0 < K; k0 += 4) {
        v2f a = (k0 < K1) ? *(const v2f*)(pA1 + k0)
                          : *(const v2f*)(pA2 + (k0 - K1));
        const int pb = (k0 >> 1) * HID + bix;       // compile-time (k0>>1)*64
        v2f b0 = sB[pb +  0];
        v2f b1 = sB[pb + 16];
        v2f b2 = sB[pb + 32];
        v2f b3 = sB[pb + 48];
        acc0 = __builtin_amdgcn_wmma_f32_16x16x4_f32(false, a, false, b0, (short)0, acc0, false, false);
        acc1 = __builtin_amdgcn_wmma_f32_16x16x4_f32(false, a, false, b1, (short)0, acc1, false, false);
        acc2 = __builtin_amdgcn_wmma_f32_16x16x4_f32(false, a, false, b2, (short)0, acc2, false, false);
        acc3 = __builtin_amdgcn_wmma_f32_16x16x4_f32(false, a, false, b3, (short)0, acc3, false, false);
    }

    // C/D layout: VGPR v -> row m0 + v + 8*half, col n*16 + col
    float bb0 = 0.f, bb1 = 0.f, bb2 = 0.f, bb3 = 0.f;
    if (bias) { bb0 = bias[col]; bb1 = bias[16 + col]; bb2 = bias[32 + col]; bb3 = bias[48 + col]; }
    #pragma unroll
    for (int v = 0; v < 8; ++v) {
        float* o = out + (size_t)(m0 + v + 8 * half) * HID;
        o[ 0 + col] = acc0[v] + bb0;
        o[16 + col] = acc1[v] + bb1;
        o[32 + col] = acc2[v] + bb2;
        o[48 + col] = acc3[v] + bb3;
    }
}

// ---------------------------------------------------------------------------
// Edge aggregation: one wave per edge; lane handles channels {2*lane, 2*lane+1}.
// agg[dst] += h[src] * dinv[src]*dinv[dst]   (agg pre-zeroed; atomics hit L2)
// ---------------------------------------------------------------------------
__global__ __launch_bounds__(256) void edge_agg(
    const int* __restrict__ src, const int* __restrict__ dst,
    const float* __restrict__ h, const float* __restrict__ dinv,
    float* __restrict__ agg)
{
    int gid  = blockIdx.x * 256 + threadIdx.x;
    int e    = gid >> 5;
    int lane = gid & 31;
    if (e >= N_EDGES) return;
    int s = src[e], d = dst[e];
    float nrm = dinv[s] * dinv[d];
    v2f hv = *(const v2f*)(h + (size_t)s * HID + 2 * lane);   // coalesced b64
    float* ad = agg + (size_t)d * HID + 2 * lane;
    atomicAdd(&ad[0], hv.x * nrm);
    atomicAdd(&ad[1], hv.y * nrm);
}

// out = relu(agg + h * dinv^2 + b)   elementwise over N_NODES x HID
__global__ __launch_bounds__(256) void combine_relu(
    const float* __restrict__ agg, const float* __restrict__ h,
    const float* __restrict__ dinv, const float* __restrict__ b,
    float* __restrict__ out)
{
    int i = blockIdx.x * 256 + threadIdx.x;
    if (i >= N_NODES * HID) return;
    int n = i >> 6, c = i & 63;
    float di = dinv[n];
    float v = agg[i] + h[i] * di * di + b[c];
    out[i] = v > 0.f ? v : 0.f;
}

// global mean pool (sums + counts via atomics; buffers pre-zeroed)
__global__ __launch_bounds__(256) void pool_sum(
    const float* __restrict__ hjk, const int* __restrict__ batch,
    float* __restrict__ pooled, float* __restrict__ cnt)
{
    int i = blockIdx.x * 256 + threadIdx.x;
    if (i >= N_NODES * HID) return;
    int n = i >> 6, c = i & 63;
    int g = batch[n];
    atomicAdd(&pooled[g * HID + c], hjk[i]);
    if (c == 0) atomicAdd(&cnt[g], 1.0f);
}

// head: logits = (pooled/cnt) @ Wfc + bfc ; softmax.  1 block, thread = graph.
__global__ __launch_bounds__(256) void head_softmax(
    const float* __restrict__ pooled, const float* __restrict__ cnt,
    const float* __restrict__ Wfc, const float* __restrict__ bfc,
    float* __restrict__ out)
{
    int g = threadIdx.x;
    float inv = 1.0f / fmaxf(cnt[g], 1.0f);
    float logit[OUT_CH];
    #pragma unroll
    for (int j = 0; j < OUT_CH; ++j) logit[j] = bfc[j];
    for (int k = 0; k < HID; ++k) {
        float p = pooled[g * HID + k] * inv;
        #pragma unroll
        for (int j = 0; j < OUT_CH; ++j) logit[j] += p * Wfc[k * OUT_CH + j];
    }
    float m = logit[0];
    #pragma unroll
    for (int j = 1; j < OUT_CH; ++j) m = fmaxf(m, logit[j]);
    float s = 0.f, e[OUT_CH];
    #pragma unroll
    for (int j = 0; j < OUT_CH; ++j) { e[j] = __expf(logit[j] - m); s += e[j]; }
    float is = 1.0f / s;
    #pragma unroll
    for (int j = 0; j < OUT_CH; ++j) out[g * OUT_CH + j] = e[j] * is;
}

// ---------------------------------------------------------------------------
extern "C" void kernel_launch(void* const* d_in, const int* in_sizes, int n_in,
                              void* d_out, int out_size, void* d_ws, size_t ws_size,
                              hipStream_t stream) {
    const float* x    = (const float*)d_in[0];
    const int*   ei   = (const int*)d_in[1];      // [2, E]: row0=src, row1=dst
    const int*   srcp = ei;
    const int*   dstp = ei + N_EDGES;
    const int*   batch= (const int*)d_in[2];
    const float* W1   = (const float*)d_in[3];
    const float* b1   = (const float*)d_in[4];
    const float* W2   = (const float*)d_in[5];
    const float* b2   = (const float*)d_in[6];
    const float* Wjk  = (const float*)d_in[7];
    const float* bjk  = (const float*)d_in[8];
    const float* Wfc  = (const float*)d_in[9];
    const float* bfc  = (const float*)d_in[10];
    float* out = (float*)d_out;

    const size_t NH = (size_t)N_NODES * HID;      // 6.4M floats
    float* ws     = (float*)d_ws;
    float* dinv   = ws;                           // N_NODES (doubles as deg)
    float* bufA   = ws + 100352;                  // GEMM output / hjk
    float* bufB   = bufA + NH;                    // agg scratch
    float* bufC   = bufB + NH;                    // h1
    float* bufD   = bufC + NH;                    // h2
    float* pooled = bufD + NH;                    // 256*64
    float* cnt    = pooled + (size_t)N_GRAPHS * HID; // 256

    const int TB = 256;
    const int gEdge  = (N_EDGES + TB - 1) / TB;              // per-edge threads
    const int gEdgeW = (N_EDGES * 32) / TB;                  // wave-per-edge
    const int gElem  = (int)((NH + TB - 1) / TB);            // per-element
    const int gGemm  = (N_NODES + 127) / 128;                // 128 rows / block

    // degrees -> dinv
    hipMemsetAsync(dinv, 0, (size_t)N_NODES * sizeof(float), stream);
    deg_count<<<gEdge, TB, 0, stream>>>(dstp, dinv);
    make_dinv<<<(N_NODES + TB - 1) / TB, TB, 0, stream>>>(dinv);

    // layer 1: h = x @ W1 ; agg ; h1 = relu(agg + h*dinv^2 + b1)
    gemm_wmma_f32<IN_CH, 0><<<gGemm, TB, 0, stream>>>(x, nullptr, W1, nullptr, bufA, N_NODES);
    hipMemsetAsync(bufB, 0, NH * sizeof(float), stream);
    edge_agg<<<gEdgeW, TB, 0, stream>>>(srcp, dstp, bufA, dinv, bufB);
    combine_relu<<<gElem, TB, 0, stream>>>(bufB, bufA, dinv, b1, bufC);

    // layer 2: h = h1 @ W2 ; agg ; h2 = relu(agg + h*dinv^2 + b2)
    gemm_wmma_f32<HID, 0><<<gGemm, TB, 0, stream>>>(bufC, nullptr, W2, nullptr, bufA, N_NODES);
    hipMemsetAsync(bufB, 0, NH * sizeof(float), stream);
    edge_agg<<<gEdgeW, TB, 0, stream>>>(srcp, dstp, bufA, dinv, bufB);
    combine_relu<<<gElem, TB, 0, stream>>>(bufB, bufA, dinv, b2, bufD);

    // JK: hjk = [h1 | h2] @ Wjk + bjk
    gemm_wmma_f32<HID, HID><<<gGemm, TB, 0, stream>>>(bufC, bufD, Wjk, bjk, bufA, N_NODES);

    // global mean pool + head
    hipMemsetAsync(pooled, 0, ((size_t)N_GRAPHS * HID + N_GRAPHS) * sizeof(float), stream);
    pool_sum<<<gElem, TB, 0, stream>>>(bufA, batch, pooled, cnt);
    head_softmax<<<1, TB, 0, stream>>>(pooled, cnt, Wfc, bfc, out);
}